// RejectionSampler_21818433864040
// MI455X (gfx1250) — compile-verified
//
#include <hip/hip_runtime.h>
#include <math.h>

#define BLOCK 256
#define DEPTH 4      // async pipeline depth (LDS buffers)
#define PRE   3      // prefetch distance (chunks in flight beyond current)

struct ShMem {
    float4 buf[DEPTH][BLOCK];   // async-load staging: 4 * 4KB = 16KB
    float  rmax[BLOCK / 32];
    int    ridx[BLOCK / 32];
};

// One block per logits row (row length `vocab` floats, vocab >= 4*BLOCK*4).
// Streams the row through LDS via CDNA5 async global->LDS b128 loads
// (4-deep pipeline on ASYNCcnt). Argmax uses chunk-max screening with a rare
// divergent update path; first-occurrence tie-break preserved throughout.
__global__ void __launch_bounds__(BLOCK)
argmax_rows_kernel(const float* __restrict__ logits, int* __restrict__ out_idx,
                   int vocab) {
    __shared__ ShMem sh;
    const int tid    = threadIdx.x;
    const int row    = blockIdx.x;
    const int vocab4 = vocab >> 2;                        // float4 per row
    const int chunks = (vocab4 + BLOCK - 1) / BLOCK;      // 125 for 128000

    const unsigned long long rowbase =
        (unsigned long long)(const void*)(logits) +
        ((unsigned long long)row * (unsigned)vocab) * 4ull;
    const unsigned lds_tid =
        (unsigned)(size_t)(&sh.buf[0][0]) + (unsigned)tid * 16u;

    // ---- issue state (incremental) ----
    unsigned long long ga = rowbase + (unsigned long long)tid * 16ull;

    // ---- process state ----
    float best  = -INFINITY;
    int   bidx  = 0x7fffffff;
    int   pbase = tid << 2;                               // element idx of comp 0

    // process one chunk already resident in LDS slot `slot`
    auto process = [&](int slot, bool guard) {
        float4 v = sh.buf[slot][tid];
        float x0 = v.x, x1 = v.y, x2 = v.z, x3 = v.w;
        if (guard) {
            if (pbase + 0 >= vocab) x0 = -INFINITY;
            if (pbase + 1 >= vocab) x1 = -INFINITY;
            if (pbase + 2 >= vocab) x2 = -INFINITY;
            if (pbase + 3 >= vocab) x3 = -INFINITY;
        }
        float cm = fmaxf(fmaxf(x0, x1), fmaxf(x2, x3));
        if (cm > best) {                                  // rare after warm-up
            best = cm;
            bidx = (x0 == cm) ? pbase
                 : (x1 == cm) ? pbase + 1
                 : (x2 == cm) ? pbase + 2
                 :              pbase + 3;                // first occurrence
        }
        pbase += BLOCK * 4;
    };

    // ---- prime: 3 chunks in flight (chunks >= 4 for this problem) ----
    #pragma unroll
    for (int k = 0; k < PRE; ++k) {
        unsigned lo = lds_tid + (unsigned)k * (BLOCK * 16u);
        asm volatile("global_load_async_to_lds_b128 %0, %1, off"
                     :: "v"(lo), "v"(ga) : "memory");
        ga += (unsigned long long)(BLOCK * 16);
    }

    // ---- steady state: issue k+3, wait for k, consume k ----
    int k = 0;
    #pragma unroll 4
    for (; k + PRE < chunks; ++k) {
        unsigned lo = lds_tid + (unsigned)((k + PRE) & (DEPTH - 1)) * (BLOCK * 16u);
        asm volatile("global_load_async_to_lds_b128 %0, %1, off"
                     :: "v"(lo), "v"(ga) : "memory");
        ga += (unsigned long long)(BLOCK * 16);
        asm volatile("s_wait_asynccnt 0x3" ::: "memory");  // chunk k landed
        process(k & (DEPTH - 1), false);
    }

    // ---- drain: 3 chunks still in flight, no new issues ----
    asm volatile("s_wait_asynccnt 0x2" ::: "memory");
    process(k & (DEPTH - 1), true); ++k;
    asm volatile("s_wait_asynccnt 0x1" ::: "memory");
    process(k & (DEPTH - 1), true); ++k;
    asm volatile("s_wait_asynccnt 0x0" ::: "memory");
    process(k & (DEPTH - 1), true);

    // ---- wave32 butterfly reduction (first-occurrence tie-break) ----
    #pragma unroll
    for (int off = 16; off > 0; off >>= 1) {
        float ov = __shfl_xor(best, off, 32);
        int   oi = __shfl_xor(bidx, off, 32);
        if (ov > best || (ov == best && oi < bidx)) { best = ov; bidx = oi; }
    }
    const int wid = tid >> 5, lane = tid & 31;
    if (lane == 0) { sh.rmax[wid] = best; sh.ridx[wid] = bidx; }
    __syncthreads();
    if (tid == 0) {
        #pragma unroll
        for (int w = 1; w < BLOCK / 32; ++w) {
            float ov = sh.rmax[w]; int oi = sh.ridx[w];
            if (ov > best || (ov == best && oi < bidx)) { best = ov; bidx = oi; }
        }
        out_idx[row] = bidx;
    }
}

// One thread per batch row: ragged gather + cumprod accept + force-emit.
__global__ void finalize_kernel(const int* __restrict__ argmax_ids,
                                const int* __restrict__ spec,
                                const int* __restrict__ spec_lens,
                                const int* __restrict__ cu_starts,
                                float* __restrict__ out,
                                int B, int max_spec) {
    int b = blockIdx.x * blockDim.x + threadIdx.x;
    if (b >= B) return;
    const int K1 = max_spec + 1;                          // 8
    const int samp  = spec_lens[b] + 1;
    const int start = cu_starts[b];

    int  toks[8];
    bool g[8];
    int acc = 1;
    for (int p = 0; p < K1 && p < 8; ++p) {
        bool valid = p < samp;
        int t = valid ? argmax_ids[start + p] : -1;       // output_token_ids (pad -1)
        toks[p] = t;
        bool gm = false;
        if (p < max_spec) {                               // draft positions
            acc *= (t == spec[b * max_spec + p]) ? 1 : 0;
            gm = (acc != 0) && valid;
        }                                                 // bonus col: always false
        g[p] = gm;
    }
    // force-emit first rejected/bonus position (col K1-1 is always false)
    int fz = K1 - 1;
    for (int p = 0; p < K1 && p < 8; ++p) { if (!g[p]) { fz = p; break; } }
    g[fz] = true;

    for (int p = 0; p < K1 && p < 8; ++p)
        out[b * K1 + p] = g[p] ? (float)toks[p] : -1.0f;
}

extern "C" void kernel_launch(void* const* d_in, const int* in_sizes, int n_in,
                              void* d_out, int out_size, void* d_ws, size_t ws_size,
                              hipStream_t stream) {
    const float* logits    = (const float*)d_in[0];
    const int*   spec      = (const int*)d_in[1];
    const int*   spec_lens = (const int*)d_in[2];
    const int*   cu_starts = (const int*)d_in[3];
    float*       out       = (float*)d_out;

    const int vocab    = 128000;                  // fixed by reference
    const int total    = in_sizes[0] / vocab;     // 635 ragged logits rows
    const int B        = in_sizes[2];             // 128
    const int max_spec = in_sizes[1] / B;         // 7

    int* d_argmax = (int*)d_ws;                   // `total` ints of scratch

    argmax_rows_kernel<<<total, BLOCK, 0, stream>>>(logits, d_argmax, vocab);
    finalize_kernel<<<(B + 127) / 128, 128, 0, stream>>>(
        d_argmax, spec, spec_lens, cu_starts, out, B, max_spec);
}